// RSNNet_7816840478842
// MI455X (gfx1250) — compile-verified
//
#include <hip/hip_runtime.h>

// Problem constants (match reference)
#define T_STEPS 16
#define B_SZ    512
#define N_DIM   4096
#define THRESH  1.0f
#define BETA    0.95f

typedef float v2f __attribute__((ext_vector_type(2)));
typedef float v8f __attribute__((ext_vector_type(8)));

#define LDSPITCH 36   // padded row stride (floats): 144B keeps float4 16B-aligned

// ---------------------------------------------------------------------------
// Tiled WMMA GEMM computing C[m][n] = sum_k A[m][k] * W[n][k]  (i.e. A @ W^T)
// Block: 128 threads = 4 waves stacked in M. Workgroup tile: 128(M) x 64(N).
// Per wave: 2 M-tiles x 4 N-tiles -> 8x v8f f32 accumulators (64 VGPRs).
// K staged by 32 through double-buffered LDS (ping-pong, reg-staged prefetch).
// MODE 0: out0[m*N+n] = acc + bias[n]                  (cur0, computed once)
// MODE 1: fused layer-1 LIF epilogue writing spk_rec[t], mem_rec[t], m1.
// ---------------------------------------------------------------------------
template <int MODE>
__global__ __launch_bounds__(128) void gemm_lif_kernel(
    const float* __restrict__ A,        // [B, N]  (x for layer0, spk0 for layer1)
    const float* __restrict__ W,        // [N, N]  row-major: W[n][k]
    const float* __restrict__ bias,     // [N]
    const float* __restrict__ vscale,   // [1] (MODE 1 only)
    float* __restrict__ out0,           // MODE 0: cur0 [B,N]; MODE 1: d_out base
    float* __restrict__ mem_state,      // MODE 1: m1 [B,N]
    const float* __restrict__ spk_in,   // MODE 1: spk0 [B,N]
    int t)
{
    __shared__ float As[2][128 * LDSPITCH];   // 2 x 18 KB
    __shared__ float Bs[2][64 * LDSPITCH];    // 2 x  9 KB  (54 KB total)

    const int tid  = threadIdx.x;
    const int lane = tid & 31;
    const int wave = tid >> 5;     // 0..3 -> M sub-block of 32 rows
    const int lm   = lane & 15;    // row within 16 (WMMA lane layout)
    const int half = lane >> 4;    // 0: K pair {0,1}, 1: K pair {2,3}

    const int row0 = blockIdx.y * 128;  // M block base (batch rows)
    const int col0 = blockIdx.x * 64;   // N block base (output features)

    v8f acc[8] = {};   // [mt*4 + nt]

    // Cooperative global->LDS staging: 128 thr, float4 granularity.
    // A tile 128x32 -> 8 float4/thread ; B tile 64x32 -> 4 float4/thread.
    const int lr = tid >> 3;          // 0..15
    const int lc = (tid & 7) * 4;     // 0,4,...,28

    const float* aBase = A + (size_t)(row0 + lr) * N_DIM + lc;
    const float* wBase = W + (size_t)(col0 + lr) * N_DIM + lc;

    float4 pa[8], pb[4];

    // ---- prologue: stage 0 into buffer 0 ----
    #pragma unroll
    for (int i = 0; i < 8; ++i) pa[i] = *(const float4*)(aBase + (size_t)i * 16 * N_DIM);
    #pragma unroll
    for (int i = 0; i < 4; ++i) pb[i] = *(const float4*)(wBase + (size_t)i * 16 * N_DIM);
    #pragma unroll
    for (int i = 0; i < 8; ++i) *(float4*)(&As[0][(lr + i * 16) * LDSPITCH + lc]) = pa[i];
    #pragma unroll
    for (int i = 0; i < 4; ++i) *(float4*)(&Bs[0][(lr + i * 16) * LDSPITCH + lc]) = pb[i];
    __syncthreads();

    const int NSTAGE = N_DIM / 32;    // 128 K-stages

    for (int s = 0; s < NSTAGE; ++s) {
        const int cur = s & 1;

        // prefetch next stage into registers (overlaps with WMMA below)
        if (s + 1 < NSTAGE) {
            const int kb = (s + 1) * 32;
            #pragma unroll
            for (int i = 0; i < 8; ++i) pa[i] = *(const float4*)(aBase + (size_t)i * 16 * N_DIM + kb);
            #pragma unroll
            for (int i = 0; i < 4; ++i) pb[i] = *(const float4*)(wBase + (size_t)i * 16 * N_DIM + kb);
        }

        // compute this stage: 8 K-pairs x (2 M-tiles x 4 N-tiles) = 64 wmma
        #pragma unroll
        for (int kk = 0; kk < 32; kk += 4) {
            const v2f a0 = *(const v2f*)(&As[cur][(wave * 32 + lm) * LDSPITCH + kk + 2 * half]);
            const v2f a1 = *(const v2f*)(&As[cur][(wave * 32 + 16 + lm) * LDSPITCH + kk + 2 * half]);
            #pragma unroll
            for (int nt = 0; nt < 4; ++nt) {
                const v2f b = *(const v2f*)(&Bs[cur][(nt * 16 + lm) * LDSPITCH + kk + 2 * half]);
                acc[nt] = __builtin_amdgcn_wmma_f32_16x16x4_f32(
                    false, a0, false, b, (short)0, acc[nt], false, false);
                acc[4 + nt] = __builtin_amdgcn_wmma_f32_16x16x4_f32(
                    false, a1, false, b, (short)0, acc[4 + nt], false, false);
            }
        }

        // store prefetched stage into the other buffer
        if (s + 1 < NSTAGE) {
            const int nxt = 1 - cur;
            #pragma unroll
            for (int i = 0; i < 8; ++i) *(float4*)(&As[nxt][(lr + i * 16) * LDSPITCH + lc]) = pa[i];
            #pragma unroll
            for (int i = 0; i < 4; ++i) *(float4*)(&Bs[nxt][(lr + i * 16) * LDSPITCH + lc]) = pb[i];
        }
        __syncthreads();
    }

    // Epilogue. C/D layout: VGPR v, lanes 0-15 -> M = v, lanes 16-31 -> M = v+8,
    // N = lm within the 16-wide tile.
    const float vs = (MODE == 1) ? vscale[0] : 0.0f;
    const size_t BN = (size_t)B_SZ * N_DIM;

    #pragma unroll
    for (int mt = 0; mt < 2; ++mt) {
        #pragma unroll
        for (int nt = 0; nt < 4; ++nt) {
            const int n  = col0 + nt * 16 + lm;
            const float bn = bias[n];
            #pragma unroll
            for (int v = 0; v < 8; ++v) {
                const int m = row0 + wave * 32 + mt * 16 + v + 8 * half;
                const size_t idx = (size_t)m * N_DIM + n;
                const float curv = acc[mt * 4 + nt][v] + bn;
                if (MODE == 0) {
                    out0[idx] = curv;
                } else {
                    const float s0    = spk_in[idx];
                    const float mp    = mem_state[idx];
                    const float reset = (mp > THRESH) ? 1.0f : 0.0f;
                    const float mn    = BETA * mp + curv + vs * s0 - reset * THRESH;
                    const float sp    = ((mn - THRESH) > 0.0f) ? 1.0f : 0.0f;
                    out0[(size_t)t * BN + idx]             = sp;  // spk_rec[t]
                    out0[(size_t)(T_STEPS + t) * BN + idx] = mn;  // mem_rec[t]
                    mem_state[idx] = mn;
                }
            }
        }
    }
}

// ---------------------------------------------------------------------------
// Layer-0 LIF: elementwise over [B, N]. cur0 is precomputed (loop-invariant).
// ---------------------------------------------------------------------------
__global__ __launch_bounds__(256) void lif0_kernel(
    const float* __restrict__ cur0, const float* __restrict__ x,
    const float* __restrict__ v0, float* __restrict__ m0,
    float* __restrict__ spk0, int n)
{
    const int i = blockIdx.x * blockDim.x + threadIdx.x;
    if (i >= n) return;
    const float mp    = m0[i];
    const float reset = (mp > THRESH) ? 1.0f : 0.0f;
    const float mn    = BETA * mp + cur0[i] + v0[0] * x[i] - reset * THRESH;
    m0[i]   = mn;
    spk0[i] = ((mn - THRESH) > 0.0f) ? 1.0f : 0.0f;
}

// ---------------------------------------------------------------------------
extern "C" void kernel_launch(void* const* d_in, const int* in_sizes, int n_in,
                              void* d_out, int out_size, void* d_ws, size_t ws_size,
                              hipStream_t stream)
{
    const float* x  = (const float*)d_in[0];   // [B, N]
    const float* W0 = (const float*)d_in[1];   // [N, N]
    const float* b0 = (const float*)d_in[2];   // [N]
    const float* v0 = (const float*)d_in[3];   // [1]
    const float* W1 = (const float*)d_in[4];   // [N, N]
    const float* b1 = (const float*)d_in[5];   // [N]
    const float* v1 = (const float*)d_in[6];   // [1]
    float* out = (float*)d_out;                // [2, T, B, N]

    const size_t BN = (size_t)B_SZ * N_DIM;
    float* cur0 = (float*)d_ws;     // [B,N]  8 MB
    float* spk0 = cur0 + BN;        // [B,N]  8 MB
    float* m0   = spk0 + BN;        // [B,N]  8 MB
    float* m1   = m0 + BN;          // [B,N]  8 MB  (32 MB total)

    hipMemsetAsync(m0, 0, BN * sizeof(float), stream);
    hipMemsetAsync(m1, 0, BN * sizeof(float), stream);

    dim3 grid(N_DIM / 64, B_SZ / 128);   // 64 x 4 = 256 workgroups
    dim3 block(128);

    // cur0 = x @ W0^T + b0 : loop-invariant across the 16 steps -> once.
    gemm_lif_kernel<0><<<grid, block, 0, stream>>>(
        x, W0, b0, nullptr, cur0, nullptr, nullptr, 0);

    for (int t = 0; t < T_STEPS; ++t) {
        lif0_kernel<<<(int)(BN / 256), 256, 0, stream>>>(cur0, x, v0, m0, spk0, (int)BN);
        gemm_lif_kernel<1><<<grid, block, 0, stream>>>(
            spk0, W1, b1, v1, out, m1, spk0, t);
    }
}